// DSMIL_20143396619264
// MI455X (gfx1250) — compile-verified
//
#include <hip/hip_runtime.h>

typedef __attribute__((ext_vector_type(2))) float v2f;
typedef __attribute__((ext_vector_type(8))) float v8f;

#define MID 512
#define CLS 2
#define ROWS3 400
#define WPB 4              // waves per block in the WMMA kernel
#define ASTRIDE 68         // LDS row stride (floats): 272B, 16B-aligned, conflict-free

// ---------------- K0: init argmax accumulators ----------------
__global__ void k0_init(unsigned long long* argmax) {
  if (threadIdx.x < CLS) argmax[threadIdx.x] = 0ull;
}

// ---------------- K1 / K3a: O[n,0..1] = feat[n,:] . W2[c,:] + bias2[c] ----
// WMMA f32 16x16x4, async global->LDS staging (ASYNCcnt), double-buffered.
__global__ __launch_bounds__(128)
void proj2_wmma(const float* __restrict__ feat,
                const float* __restrict__ W2,     // [2,512] row-major
                const float* __restrict__ bias2,  // [2]
                float* __restrict__ out2,         // [N,2]
                int N,
                unsigned long long* argmax)       // null -> no argmax epilogue
{
  __shared__ float sW[CLS * MID];
  __shared__ float sA[WPB][2][16 * ASTRIDE];      // double-buffered 16x64 tiles

  const int tid  = threadIdx.x;
  const int wave = tid >> 5;
  const int lane = tid & 31;
  const int jcol = lane & 15;                     // B column / A row index
  const int half = lane >> 4;

  for (int i = tid; i < CLS * MID; i += 128) sW[i] = W2[i];
  __syncthreads();                                // sW visible to all waves

  const int rowBase = blockIdx.x * (WPB * 16) + wave * 16;

  // Per-lane staging: slot t2 covers row = half + 2*t2, float4 column jcol.
  const float* gp[8];
  unsigned ldsOff[8];
#pragma unroll
  for (int t2 = 0; t2 < 8; ++t2) {
    const int r = half + 2 * t2;
    int grow = rowBase + r; if (grow > N - 1) grow = N - 1;
    gp[t2]     = feat + (size_t)grow * MID + jcol * 4;
    ldsOff[t2] = (unsigned)((r * ASTRIDE + jcol * 4) * 4);   // 16B aligned
  }
  const unsigned bufOff0 = (unsigned)(unsigned long long)(const void*)&sA[wave][0][0];
  const unsigned bufOff1 = (unsigned)(unsigned long long)(const void*)&sA[wave][1][0];

  auto issueChunk = [&](int kc, unsigned bufOff) {
    const int base = kc * 64;
#pragma unroll
    for (int t2 = 0; t2 < 8; ++t2) {
      asm volatile("global_load_async_to_lds_b128 %0, %1, off"
                   :: "v"(bufOff + ldsOff[t2]), "v"(gp[t2] + base)
                   : "memory");
    }
  };

  v8f acc = {0.f, 0.f, 0.f, 0.f, 0.f, 0.f, 0.f, 0.f};
  // Unused D columns (jcol >= CLS) are never read, and WMMA columns are
  // independent, so alias them onto a real class row: no masking needed.
  const int js = jcol & 1;

  issueChunk(0, bufOff0);
  for (int kc = 0; kc < MID / 64; ++kc) {
    if (kc < MID / 64 - 1) {
      issueChunk(kc + 1, (kc & 1) ? bufOff0 : bufOff1);
      asm volatile("s_wait_asynccnt 0x8" ::: "memory");  // chunk kc landed
    } else {
      asm volatile("s_wait_asynccnt 0x0" ::: "memory");
    }
    const float* sa = &sA[wave][kc & 1][0];
    const int base = kc * 64;
#pragma unroll
    for (int j = 0; j < 16; ++j) {                // 16 x (K=4) steps
      const int kk = j * 4 + half * 2;            // local K pair for this lane
      v2f a, b;
      a.x = sa[jcol * ASTRIDE + kk];              // A[m=jcol][K=kk]
      a.y = sa[jcol * ASTRIDE + kk + 1];
      const int kg = base + kk;
      b.x = sW[js * MID + kg];                    // B[k][jcol] = W2[jcol][k]
      b.y = sW[js * MID + kg + 1];
      acc = __builtin_amdgcn_wmma_f32_16x16x4_f32(false, a, false, b,
                                                  (short)0, acc, false, false);
    }
    asm volatile("s_wait_dscnt 0x0" ::: "memory"); // tile reads done before reuse
  }

  // Epilogue: C/D layout -> lane l vgpr r holds D[M = r + 8*half][N = jcol]
  const float bj = (jcol < CLS) ? bias2[jcol] : 0.f;
  float bestv = -3.4e38f;
  int   bestn = 0;
#pragma unroll
  for (int r = 0; r < 8; ++r) {
    const int grow = rowBase + r + half * 8;
    const float val = acc[r] + bj;
    if (jcol < CLS && grow < N) {
      out2[(size_t)grow * CLS + jcol] = val;
      if (val > bestv) { bestv = val; bestn = grow; }   // strict > keeps first
    }
  }
  if (argmax && jcol < CLS) {
    unsigned uv = __float_as_uint(bestv);
    uv = (uv & 0x80000000u) ? ~uv : (uv | 0x80000000u); // order-preserving map
    unsigned long long key =
        ((unsigned long long)uv << 32) |
        (unsigned long long)(0xFFFFFFFFu - (unsigned)bestn); // tie -> smallest n
    atomicMax(&argmax[jcol], key);
  }
}

// ---------------- K2: q_max = feat[idx] @ w_q^T + b_q ; u = (w_q^T q_max)/sqrt(D)
__global__ __launch_bounds__(512)
void k2_critical(const float* __restrict__ feat, const float* __restrict__ w_q,
                 const float* __restrict__ b_q,
                 const unsigned long long* __restrict__ argmax,
                 float* __restrict__ u, float* __restrict__ s)
{
  __shared__ float qm[CLS][MID];
  __shared__ int sidx[CLS];
  const int t = threadIdx.x;
  if (t < CLS) sidx[t] = (int)(0xFFFFFFFFu - (unsigned)(argmax[t] & 0xFFFFFFFFull));
  __syncthreads();
  for (int c = 0; c < CLS; ++c) {
    const float* fr = feat + (size_t)sidx[c] * MID;
    float a0 = 0.f;
    for (int k = 0; k < MID; ++k) a0 = fmaf(fr[k], w_q[(size_t)t * MID + k], a0);
    qm[c][t] = a0 + b_q[t];
  }
  __syncthreads();
  const float invs = 0.04419417382415922f;        // 1/sqrt(512)
  for (int c = 0; c < CLS; ++c) {
    float a0 = 0.f;
    for (int d = 0; d < MID; ++d) a0 = fmaf(w_q[(size_t)d * MID + t], qm[c][d], a0);
    u[c * MID + t] = a0 * invs;
  }
  if (t < CLS) {
    float a0 = 0.f;
    for (int d = 0; d < MID; ++d) a0 = fmaf(b_q[d], qm[t][d], a0);
    s[t] = a0 * invs;
  }
}

// ---------------- K3r: per-class max & 1/sum(exp) over N (deterministic) ----
__global__ __launch_bounds__(1024)
void k3r_stats(const float* __restrict__ logits, int N,
               float* __restrict__ maxv, float* __restrict__ invsum)
{
  __shared__ float red[1024];
  const int t = threadIdx.x;
  for (int c = 0; c < CLS; ++c) {
    float m = -3.4e38f;
    for (int n = t; n < N; n += 1024) m = fmaxf(m, logits[(size_t)n * CLS + c]);
    red[t] = m; __syncthreads();
    for (int o = 512; o > 0; o >>= 1) { if (t < o) red[t] = fmaxf(red[t], red[t + o]); __syncthreads(); }
    const float M = red[0]; __syncthreads();
    float sum = 0.f;
    for (int n = t; n < N; n += 1024) sum += expf(logits[(size_t)n * CLS + c] - M);
    red[t] = sum; __syncthreads();
    for (int o = 512; o > 0; o >>= 1) { if (t < o) red[t] += red[t + o]; __syncthreads(); }
    if (t == 0) { maxv[c] = M; invsum[c] = 1.f / red[0]; }
    __syncthreads();
  }
}

// ---------------- K3b: per-block partial of Sum_n e[n,c] * feat[n,:] --------
__global__ __launch_bounds__(256)
void k3b_wsum(const float* __restrict__ feat, const float* __restrict__ logits,
              const float* __restrict__ maxv, int N, float* __restrict__ part)
{
  const int t = threadIdx.x;
  const int b = blockIdx.x;
  const int n0 = b * ROWS3;
  const int n1 = (n0 + ROWS3 < N) ? n0 + ROWS3 : N;
  const float m0 = maxv[0], m1 = maxv[1];
  const int d = 2 * t;                            // float2 lane per thread
  float a00 = 0.f, a01 = 0.f, a10 = 0.f, a11 = 0.f;
  for (int n = n0; n < n1; ++n) {
    const float e0 = expf(logits[(size_t)n * CLS + 0] - m0);
    const float e1 = expf(logits[(size_t)n * CLS + 1] - m1);
    const float2 f = *(const float2*)(feat + (size_t)n * MID + d);
    a00 = fmaf(e0, f.x, a00); a01 = fmaf(e1, f.x, a01);
    a10 = fmaf(e0, f.y, a10); a11 = fmaf(e1, f.y, a11);
  }
  float* p = part + (size_t)b * (CLS * MID);
  p[0 * MID + d]     = a00; p[1 * MID + d]     = a01;
  p[0 * MID + d + 1] = a10; p[1 * MID + d + 1] = a11;
}

// ---------------- K4: F[c,d] = (Sum_b part) * invsum[c] ---------------------
__global__ void k4_reduceF(const float* __restrict__ part,
                           const float* __restrict__ invsum,
                           float* __restrict__ F, int nb)
{
  const int i = blockIdx.x * 256 + threadIdx.x;   // 0..1023
  if (i < CLS * MID) {
    float acc = 0.f;
    for (int b = 0; b < nb; ++b) acc += part[(size_t)b * (CLS * MID) + i];
    F[i] = acc * invsum[i >> 9];
  }
}

// ---------------- K5: bag B[c,d] = F[c,:].w_v[d,:] + b_v[d] -----------------
__global__ void k5_bag(const float* __restrict__ F, const float* __restrict__ w_v,
                       const float* __restrict__ b_v, float* __restrict__ outbag)
{
  const int d = blockIdx.x * 128 + threadIdx.x;   // 0..511
  if (d < MID) {
    float a0 = 0.f, a1 = 0.f;
    for (int k = 0; k < MID; ++k) {
      const float w = w_v[(size_t)d * MID + k];
      a0 = fmaf(F[k], w, a0);
      a1 = fmaf(F[MID + k], w, a1);
    }
    outbag[d]       = a0 + b_v[d];
    outbag[MID + d] = a1 + b_v[d];
  }
}

// ---------------- K6: C_out[o] = Sum_{c,d} B[c,d] w_conv[o,c,d] + b_conv ----
__global__ __launch_bounds__(256)
void k6_conv(const float* __restrict__ bag, const float* __restrict__ w_conv,
             const float* __restrict__ b_conv, float* __restrict__ outC)
{
  __shared__ float red[256];
  const int t = threadIdx.x;
  for (int o = 0; o < CLS; ++o) {
    float acc = 0.f;
    for (int i = t; i < CLS * MID; i += 256)
      acc = fmaf(bag[i], w_conv[o * CLS * MID + i], acc);
    red[t] = acc; __syncthreads();
    for (int s2 = 128; s2 > 0; s2 >>= 1) { if (t < s2) red[t] += red[t + s2]; __syncthreads(); }
    if (t == 0) outC[o] = red[0] + b_conv[o];
    __syncthreads();
  }
}

extern "C" void kernel_launch(void* const* d_in, const int* in_sizes, int n_in,
                              void* d_out, int out_size, void* d_ws, size_t ws_size,
                              hipStream_t stream)
{
  const float* x      = (const float*)d_in[0];
  const float* w_fc   = (const float*)d_in[1];
  const float* b_fc   = (const float*)d_in[2];
  const float* w_q    = (const float*)d_in[3];
  const float* b_q    = (const float*)d_in[4];
  const float* w_v    = (const float*)d_in[5];
  const float* b_v    = (const float*)d_in[6];
  const float* w_conv = (const float*)d_in[7];
  const float* b_conv = (const float*)d_in[8];
  const int N = in_sizes[0] / MID;

  float* out     = (float*)d_out;
  float* out_C   = out;                   // [1,2]
  float* out_bag = out + CLS;             // [1, CLS*MID]
  float* out_ip  = out + CLS + CLS * MID; // [N,2]

  char* ws = (char*)d_ws;
  unsigned long long* argmax = (unsigned long long*)(ws);          // 16 B
  float* u      = (float*)(ws + 16);                                // 4096 B
  float* s      = (float*)(ws + 16 + 4096);                         // 8 B
  float* maxv   = (float*)(ws + 16 + 4096 + 16);                    // 8 B
  float* invsum = (float*)(ws + 16 + 4096 + 32);                    // 8 B
  float* logits = (float*)(ws + 16 + 4096 + 64);                    // N*2*4 B
  float* F      = (float*)(ws + 16 + 4096 + 64 + (size_t)N * CLS * 4);
  float* part   = (float*)(ws + 16 + 4096 + 64 + (size_t)N * CLS * 4 + 4096);
  const int NB3 = (N + ROWS3 - 1) / ROWS3;

  const int grid1 = (N + WPB * 16 - 1) / (WPB * 16);  // 64 rows / 128-thread block

  k0_init<<<1, 32, 0, stream>>>(argmax);
  // Pass 1: inst_pred (written straight to d_out) + per-class argmax
  proj2_wmma<<<grid1, 128, 0, stream>>>(x, w_fc, b_fc, out_ip, N, argmax);
  // Tiny: q_max and the folded attention projection u, s (scale folded in)
  k2_critical<<<1, 512, 0, stream>>>(x, w_q, b_q, argmax, u, s);
  // Pass 2: attention logits (= A/sqrt(D)) via the same WMMA kernel
  proj2_wmma<<<grid1, 128, 0, stream>>>(x, u, s, logits, N, nullptr);
  // Softmax statistics (deterministic single-block reduction)
  k3r_stats<<<1, 1024, 0, stream>>>(logits, N, maxv, invsum);
  // Pass 3: weighted instance-feature sum, per-block partials then reduce
  k3b_wsum<<<NB3, 256, 0, stream>>>(x, logits, maxv, N, part);
  k4_reduceF<<<(CLS * MID + 255) / 256, 256, 0, stream>>>(part, invsum, F, NB3);
  // B = F @ w_v^T + b_v  (bag_feature), then the conv contraction
  k5_bag<<<(MID + 127) / 128, 128, 0, stream>>>(F, w_v, b_v, out_bag);
  k6_conv<<<1, 256, 0, stream>>>(out_bag, w_conv, b_conv, out_C);
}